// NumDualDescriptorRN_68470368633359
// MI455X (gfx1250) — compile-verified
//
#include <hip/hip_runtime.h>

typedef __attribute__((ext_vector_type(2))) float v2f;
typedef __attribute__((ext_vector_type(8))) float v8f;

#define L_DIM 1024
#define M_DIM 64

// ---------------------------------------------------------------------------
// Kernel 1: WT = Bbasis(1024x64) @ M^T(64x64)  ->  WT[1024x64]
// One wave32 per 16x16 output tile; K=64 accumulated via 16 chained
// v_wmma_f32_16x16x4_f32 (full fp32 precision).
// Tiles: 1024/16 * 64/16 = 64*4 = 256 waves => grid 32 blocks x 8 waves.
// ---------------------------------------------------------------------------
__global__ void __launch_bounds__(256)
wt_gemm_wmma(const float* __restrict__ Bbasis,
             const float* __restrict__ Mmat,
             float* __restrict__ WT) {
    const int wave = blockIdx.x * (blockDim.x >> 5) + (threadIdx.x >> 5);
    const int lane = threadIdx.x & 31;
    const int j0 = (wave >> 2) * 16;   // row tile in WT (j dimension)
    const int k0 = (wave & 3) * 16;    // col tile in WT (k dimension)
    const int r  = lane & 15;          // row (A) / col (B,D) within tile
    const int hi = lane >> 4;          // lane half selects K pair {0,1} vs {2,3}

    v8f acc = {};
    #pragma unroll
    for (int n = 0; n < M_DIM; n += 4) {
        // A tile: Bbasis[j0+r, n + 2*hi + {0,1}]   (16x4, MxK)
        const float* ap = Bbasis + (j0 + r) * M_DIM + n + 2 * hi;
        // B tile: (M^T)[n + 2*hi + {0,1}, k0+r] = M[k0+r, n + 2*hi + {0,1}] (4x16, KxN)
        const float* bp = Mmat   + (k0 + r) * M_DIM + n + 2 * hi;
        v2f a, b;
        a.x = ap[0]; a.y = ap[1];
        b.x = bp[0]; b.y = bp[1];
        // 8 args: (neg_a, A, neg_b, B, c_mod, C, reuse_a, reuse_b)
        acc = __builtin_amdgcn_wmma_f32_16x16x4_f32(
            false, a, false, b, (short)0, acc, false, false);
    }
    // D layout: VGPR v -> row v (lanes 0-15) / row v+8 (lanes 16-31), col = lane%16
    #pragma unroll
    for (int v = 0; v < 8; ++v) {
        WT[(j0 + v + 8 * hi) * M_DIM + k0 + r] = acc[v];
    }
}

// ---------------------------------------------------------------------------
// Kernel 2: AT[j, m] = Acoeff[m, j]  (64x1024 -> 1024x64), 256 KB one-shot.
// ---------------------------------------------------------------------------
__global__ void at_transpose(const float* __restrict__ Acoeff,
                             float* __restrict__ AT) {
    const int t = blockIdx.x * blockDim.x + threadIdx.x;   // 0 .. 65535
    const int m = t >> 10;
    const int j = t & (L_DIM - 1);
    AT[j * M_DIM + m] = Acoeff[t];
}

// ---------------------------------------------------------------------------
// Kernel 3 (bandwidth-bound main): per batch row b,
//   j = k[b] mod 1024
//   scalar = V[b,:] . WT[j,:]          (length-64 dot, 16 lanes x float4)
//   out[b,:] = scalar * AT[j,:]
// 16 lanes cooperate per row; 16 rows per 256-thread block; every global
// access is a coalesced float4 (512 B contiguous per wave per instruction).
// WT/AT (512 KB) stay resident in L2. HBM traffic ~1.03 GB => ~44 us floor.
// ---------------------------------------------------------------------------
__global__ void __launch_bounds__(256)
fused_main(const float* __restrict__ V,
           const int*   __restrict__ K,
           const float* __restrict__ WT,
           const float* __restrict__ AT,
           float*       __restrict__ out,
           int batch) {
    const int grp = threadIdx.x >> 4;                 // row group in block: 0..15
    const int sub = threadIdx.x & 15;                 // lane within group: 0..15
    const int rb  = blockIdx.x * 16 + grp;            // batch row
    if (rb >= batch) return;

    const int k = K[rb];
    const int j = ((k % L_DIM) + L_DIM) & (L_DIM - 1);

    const float4 v = ((const float4*)(V  + (size_t)rb * M_DIM))[sub];
    const float4 w = ((const float4*)(WT + (size_t)j  * M_DIM))[sub];
    float p = v.x * w.x + v.y * w.y + v.z * w.z + v.w * w.w;

    // reduce the 16-lane group (wave32: xor masks 1,2,4,8 stay within halves)
    p += __shfl_xor(p, 1);
    p += __shfl_xor(p, 2);
    p += __shfl_xor(p, 4);
    p += __shfl_xor(p, 8);

    const float4 a = ((const float4*)(AT + (size_t)j * M_DIM))[sub];
    float4 o;
    o.x = p * a.x; o.y = p * a.y; o.z = p * a.z; o.w = p * a.w;
    ((float4*)(out + (size_t)rb * M_DIM))[sub] = o;
}

// ---------------------------------------------------------------------------
extern "C" void kernel_launch(void* const* d_in, const int* in_sizes, int n_in,
                              void* d_out, int out_size, void* d_ws, size_t ws_size,
                              hipStream_t stream) {
    const float* V      = (const float*)d_in[0];   // [B, 64]
    const int*   K      = (const int*)  d_in[1];   // [B]
    const float* Mmat   = (const float*)d_in[2];   // [64, 64]
    const float* Acoeff = (const float*)d_in[3];   // [64, 1024]
    const float* Bbasis = (const float*)d_in[4];   // [1024, 64]
    float*       out    = (float*)d_out;           // [B, 64]

    const int batch = in_sizes[0] / M_DIM;

    float* WT = (float*)d_ws;                      // [1024, 64] = 256 KB
    float* AT = WT + L_DIM * M_DIM;                // [1024, 64] = 256 KB

    // Precompute (tiny): WT = Bbasis @ M^T via WMMA, AT = Acoeff^T
    wt_gemm_wmma<<<32, 256, 0, stream>>>(Bbasis, Mmat, WT);
    at_transpose<<<(L_DIM * M_DIM) / 256, 256, 0, stream>>>(Acoeff, AT);

    // Main streaming pass: one 16-lane group per batch row
    fused_main<<<(batch + 15) / 16, 256, 0, stream>>>(V, K, WT, AT, out, batch);
}